// GCN_87952340288037
// MI455X (gfx1250) — compile-verified
//
#include <hip/hip_runtime.h>

typedef __attribute__((ext_vector_type(2))) float v2f;
typedef __attribute__((ext_vector_type(8))) float v8f;

#define NN 100000
#define FDIM 128
#define SLOPE 0.03f
#define BN_EPS 1e-5f

// ---------------- utility ----------------
__global__ void fill_kernel(float* __restrict__ p, int n, float v) {
  int i = blockIdx.x * blockDim.x + threadIdx.x;
  if (i < n) p[i] = v;
}

// deg[dst] += 1 per edge (deg pre-filled with 1.0 for the self-loop)
__global__ void deg_kernel(const int* __restrict__ ei, float* __restrict__ deg, int E) {
  int e = blockIdx.x * blockDim.x + threadIdx.x;
  if (e < E) atomicAdd(&deg[ei[E + e]], 1.0f);
}

__global__ void dinv_kernel(float* __restrict__ d, int n) {
  int i = blockIdx.x * blockDim.x + threadIdx.x;
  if (i < n) d[i] = rsqrtf(d[i]);  // deg >= 1 always (self-loops)
}

// ---------------- GEMM: Out[N,128] = A[N,K] @ W[K,128] via f32 WMMA ----------------
// block = 256 threads (8 waves). Block covers 16 rows; wave w covers cols [16w,16w+16).
template <int K>
__global__ __launch_bounds__(256) void gemm_wmma(const float* __restrict__ A,
                                                 const float* __restrict__ W,
                                                 float* __restrict__ Out, int nrows) {
  __shared__ float sA[16 * K];      // 16 rows x K  (<= 8 KB)
  __shared__ float sW[64 * FDIM];   // 64-row slab of W (32 KB)
  const int tid = threadIdx.x;
  const int row0 = blockIdx.x * 16;

  for (int i = tid; i < 16 * K; i += 256) {
    int r = i / K, c = i % K;
    int gr = row0 + r;
    sA[i] = (gr < nrows) ? A[(long long)gr * K + c] : 0.0f;
  }

  const int lane = tid & 31;
  const int wave = tid >> 5;
  const int col0 = wave * 16;
  const int m  = lane & 15;         // A row within tile / B column
  const int kb = (lane >> 4) * 2;   // K sub-offset for this half-wave

  v8f acc = {0.f, 0.f, 0.f, 0.f, 0.f, 0.f, 0.f, 0.f};

  for (int kh = 0; kh < K; kh += 64) {
    __syncthreads();  // protect sW reuse (and sA on first pass)
    for (int i = tid; i < 64 * FDIM; i += 256)
      sW[i] = W[(long long)(kh + (i >> 7)) * FDIM + (i & 127)];
    __syncthreads();

    for (int k0 = 0; k0 < 64; k0 += 4) {
      v2f a, b;
      a[0] = sA[m * K + kh + k0 + kb];
      a[1] = sA[m * K + kh + k0 + kb + 1];
      b[0] = sW[(k0 + kb) * FDIM + col0 + m];
      b[1] = sW[(k0 + kb + 1) * FDIM + col0 + m];
      acc = __builtin_amdgcn_wmma_f32_16x16x4_f32(false, a, false, b,
                                                  (short)0, acc, false, false);
    }
  }

  const int mbase = (lane >> 4) * 8;
  const int n = lane & 15;
  for (int r = 0; r < 8; ++r) {
    int gr = row0 + mbase + r;
    if (gr < nrows) Out[(long long)gr * FDIM + col0 + n] = acc[r];
  }
}

// out[i] = tmp[i]*dinv[node]^2 + bias[f]   (self-loop contribution + zero-init + bias)
__global__ void self_init_kernel(const float* __restrict__ tmp, const float* __restrict__ dinv,
                                 const float* __restrict__ bias, float* __restrict__ out,
                                 int nrows) {
  int i = blockIdx.x * blockDim.x + threadIdx.x;
  if (i >= nrows * FDIM) return;
  int node = i >> 7;
  int f = i & 127;
  float dv = dinv[node];
  out[i] = tmp[i] * dv * dv + bias[f];
}

// one wave per edge: 32 lanes x float4 = 128 feats; out[dst] += tmp[src]*norm
__global__ void edge_agg_kernel(const float* __restrict__ tmp, const int* __restrict__ ei,
                                const float* __restrict__ dinv, float* __restrict__ out, int E) {
  int tid = blockIdx.x * blockDim.x + threadIdx.x;
  int e = tid >> 5;
  int c = tid & 31;
  if (e >= E) return;
  int s = ei[e];
  int d = ei[E + e];
  float norm = dinv[s] * dinv[d];
  const float4* src = (const float4*)(tmp + (long long)s * FDIM);
  float4 v = src[c];
  float* o = out + (long long)d * FDIM + c * 4;
  atomicAdd(o + 0, v.x * norm);
  atomicAdd(o + 1, v.y * norm);
  atomicAdd(o + 2, v.z * norm);
  atomicAdd(o + 3, v.w * norm);
}

// ---------------- BatchNorm (training stats) + LeakyReLU ----------------
__global__ void bn_stats_kernel(const float* __restrict__ h, float* __restrict__ sum,
                                float* __restrict__ sumsq, int N) {
  int f = threadIdx.x;  // 128 threads
  float s = 0.f, s2 = 0.f;
  for (int n = blockIdx.x; n < N; n += gridDim.x) {
    float v = h[(long long)n * FDIM + f];
    s += v;
    s2 += v * v;
  }
  atomicAdd(&sum[f], s);
  atomicAdd(&sumsq[f], s2);
}

__global__ void bn_apply_kernel(float* __restrict__ h, const float* __restrict__ sum,
                                const float* __restrict__ sumsq, const float* __restrict__ g,
                                const float* __restrict__ beta, int N) {
  int i = blockIdx.x * blockDim.x + threadIdx.x;
  if (i >= N * FDIM) return;
  int f = i & 127;
  float invn = 1.0f / (float)N;
  float mu = sum[f] * invn;
  float var = sumsq[f] * invn - mu * mu;
  float r = rsqrtf(var + BN_EPS);
  float v = (h[i] - mu) * r * g[f] + beta[f];
  h[i] = v > 0.f ? v : SLOPE * v;
}

__global__ void lrelu_kernel(float* __restrict__ h, int n) {
  int i = blockIdx.x * blockDim.x + threadIdx.x;
  if (i >= n) return;
  float v = h[i];
  h[i] = v > 0.f ? v : SLOPE * v;
}

// ---------------- pooling + dense head ----------------
__global__ void pool_cnt_kernel(const int* __restrict__ batch, float* __restrict__ cnt, int N) {
  int n = blockIdx.x * blockDim.x + threadIdx.x;
  if (n < N) atomicAdd(&cnt[batch[n]], 1.0f);
}

__global__ void pool_sum_kernel(const float* __restrict__ h, const int* __restrict__ batch,
                                float* __restrict__ ps, int N) {
  int tid = blockIdx.x * blockDim.x + threadIdx.x;
  int n = tid >> 5;
  int c = tid & 31;
  if (n >= N) return;
  int g = batch[n];
  const float4* src = (const float4*)(h + (long long)n * FDIM);
  float4 v = src[c];
  float* o = ps + (long long)g * FDIM + c * 4;
  atomicAdd(o + 0, v.x);
  atomicAdd(o + 1, v.y);
  atomicAdd(o + 2, v.z);
  atomicAdd(o + 3, v.w);
}

// out[g,o] = lin_b[o] + (ps[g,:]/max(cnt,1)) . lin_w[:,o]     (128 graphs x 64 out)
__global__ void final_dense_kernel(const float* __restrict__ ps, const float* __restrict__ cnt,
                                   const float* __restrict__ lw, const float* __restrict__ lb,
                                   float* __restrict__ out) {
  int i = blockIdx.x * blockDim.x + threadIdx.x;
  if (i >= 128 * 64) return;
  int g = i >> 6;
  int o = i & 63;
  float inv = 1.0f / fmaxf(cnt[g], 1.0f);
  float acc = lb[o];
  for (int k = 0; k < FDIM; ++k) acc += ps[g * FDIM + k] * inv * lw[k * 64 + o];
  out[i] = acc;
}

// ---------------- driver ----------------
extern "C" void kernel_launch(void* const* d_in, const int* in_sizes, int n_in,
                              void* d_out, int out_size, void* d_ws, size_t ws_size,
                              hipStream_t stream) {
  const float* x     = (const float*)d_in[0];
  const int*   ei    = (const int*)d_in[1];
  const int*   batch = (const int*)d_in[2];
  const float* W0 = (const float*)d_in[3];
  const float* b0 = (const float*)d_in[4];
  const float* W1 = (const float*)d_in[5];
  const float* b1 = (const float*)d_in[6];
  const float* W2 = (const float*)d_in[7];
  const float* b2 = (const float*)d_in[8];
  const float* g0    = (const float*)d_in[9];
  const float* beta0 = (const float*)d_in[10];
  const float* g1    = (const float*)d_in[11];
  const float* beta1 = (const float*)d_in[12];
  const float* lw = (const float*)d_in[13];
  const float* lb = (const float*)d_in[14];
  float* out = (float*)d_out;

  const int N = NN;
  const int E = in_sizes[1] / 2;

  float* ws    = (float*)d_ws;
  float* dinv  = ws;                    // N floats (deg -> dinv in place)
  float* bnsum = ws + 100096;           // 128
  float* bnsq  = bnsum + 128;           // 128
  float* cnt   = bnsq + 128;            // 128
  float* psum  = cnt + 128;             // 128*128
  float* bufT  = ws + 116864;           // N*128  (GEMM result)
  float* bufH  = bufT + (size_t)N * FDIM; // N*128 (aggregated features)

  dim3 b256(256);
  const int gemm_grid = (N + 15) / 16;
  const int nf = N * FDIM;
  const int nf_grid = (nf + 255) / 256;
  const int eg = (E * 32 + 255) / 256;

  // degree -> dinv (self-loop included via init to 1.0)
  fill_kernel<<<(N + 255) / 256, b256, 0, stream>>>(dinv, N, 1.0f);
  deg_kernel<<<(E + 255) / 256, b256, 0, stream>>>(ei, dinv, E);
  dinv_kernel<<<(N + 255) / 256, b256, 0, stream>>>(dinv, N);

  // ---- layer 0: 64 -> 128, BN0, lrelu
  gemm_wmma<64><<<gemm_grid, b256, 0, stream>>>(x, W0, bufT, N);
  self_init_kernel<<<nf_grid, b256, 0, stream>>>(bufT, dinv, b0, bufH, N);
  edge_agg_kernel<<<eg, b256, 0, stream>>>(bufT, ei, dinv, bufH, E);
  fill_kernel<<<1, b256, 0, stream>>>(bnsum, 256, 0.0f);
  bn_stats_kernel<<<1024, dim3(128), 0, stream>>>(bufH, bnsum, bnsq, N);
  bn_apply_kernel<<<nf_grid, b256, 0, stream>>>(bufH, bnsum, bnsq, g0, beta0, N);

  // ---- layer 1: 128 -> 128, BN1, lrelu
  gemm_wmma<128><<<gemm_grid, b256, 0, stream>>>(bufH, W1, bufT, N);
  self_init_kernel<<<nf_grid, b256, 0, stream>>>(bufT, dinv, b1, bufH, N);
  edge_agg_kernel<<<eg, b256, 0, stream>>>(bufT, ei, dinv, bufH, E);
  fill_kernel<<<1, b256, 0, stream>>>(bnsum, 256, 0.0f);
  bn_stats_kernel<<<1024, dim3(128), 0, stream>>>(bufH, bnsum, bnsq, N);
  bn_apply_kernel<<<nf_grid, b256, 0, stream>>>(bufH, bnsum, bnsq, g1, beta1, N);

  // ---- layer 2: 128 -> 128, lrelu only
  gemm_wmma<128><<<gemm_grid, b256, 0, stream>>>(bufH, W2, bufT, N);
  self_init_kernel<<<nf_grid, b256, 0, stream>>>(bufT, dinv, b2, bufH, N);
  edge_agg_kernel<<<eg, b256, 0, stream>>>(bufT, ei, dinv, bufH, E);
  lrelu_kernel<<<nf_grid, b256, 0, stream>>>(bufH, nf);

  // ---- mean pool + dense head
  fill_kernel<<<(128 + 128 * 128 + 255) / 256, b256, 0, stream>>>(cnt, 128 + 128 * 128, 0.0f);
  pool_cnt_kernel<<<(N + 255) / 256, b256, 0, stream>>>(batch, cnt, N);
  pool_sum_kernel<<<(N * 32 + 255) / 256, b256, 0, stream>>>(bufH, batch, psum, N);
  final_dense_kernel<<<(128 * 64 + 255) / 256, b256, 0, stream>>>(psum, cnt, lw, lb, out);
}